// LookupKAN2D_PyTorch_18657337934155
// MI455X (gfx1250) — compile-verified
//
#include <hip/hip_runtime.h>
#include <hip/hip_bf16.h>
#include <math.h>

// Problem constants
#define BATCH   16384
#define IDIM    128
#define ODIM    128
#define PDIM    64          // IDIM/2
#define G       64          // NUM_GRIDS
#define GP1     65
#define GG      4225        // 65*65
#define ROWS    540800      // GG * ODIM  (per-p plane size in floats)

typedef float v2f __attribute__((ext_vector_type(2)));
typedef float v8f __attribute__((ext_vector_type(8)));

// ---------------------------------------------------------------------------
// Kernel 1: transpose fp[i1][i2][o][p]  ->  fpT[p][g][o]   (g = i1*65+i2)
// 2-D transpose of [R=540800][64] -> [64][540800], LDS-tiled, both sides
// coalesced.  ~280 MB streaming HBM traffic (~12 us at 23.3 TB/s), once.
// ---------------------------------------------------------------------------
__global__ __launch_bounds__(256) void transpose_fp(const float* __restrict__ fp,
                                                    float* __restrict__ fpT) {
    __shared__ float tile[64][65];           // +1 pad: conflict-free
    const int r0 = blockIdx.x * 64;          // 8450 blocks cover 540800 rows
    const int t  = threadIdx.x;

    #pragma unroll
    for (int i = 0; i < 16; ++i) {
        int flat = t + 256 * i;              // 4096 elements
        int rl = flat >> 6;
        int pl = flat & 63;                  // contiguous per lane -> coalesced read
        tile[rl][pl] = fp[(size_t)(r0 + rl) * 64 + pl];
    }
    __syncthreads();
    #pragma unroll
    for (int i = 0; i < 16; ++i) {
        int flat = t + 256 * i;
        int rl = flat & 63;                  // contiguous per lane -> coalesced write
        int pl = flat >> 6;
        fpT[(size_t)pl * ROWS + r0 + rl] = tile[rl][pl];
    }
}

// ---------------------------------------------------------------------------
// Kernel 2: LookupKAN2D via V_WMMA_F32_16X16X4_F32.
// Block = 256 thr (8 waves) = one 16-batch tile; wave w owns outputs
// [16w,16w+16). Per p, per batch row b0:  D(16x16) += A(16x4) x B(4x16),
// A one-hot row b0 = bilinear weights, B = 4 gathered corner rows.
// Corner -> (VGPR,half) map identical for A and B:
//   (v0,h0)=c00  (v1,h0)=c01  (v0,h1)=c10  (v1,h1)=c11
// Params are bulk-fetched into registers per p so the unrolled 16-WMMA body
// has zero LDS ops and pure-register A construction (no branch sinking).
// ---------------------------------------------------------------------------
__global__ __launch_bounds__(256) void kan2d_wmma(const float* __restrict__ x,
                                                  const float* __restrict__ fpT,
                                                  const float* __restrict__ borders,
                                                  const float* __restrict__ invlen,
                                                  float* __restrict__ out) {
    __shared__ int    gLDS[PDIM][16];        // g00 = i1*65+i2 per (p,b)
    __shared__ float2 wLDS[PDIM][2][16];     // [half==0]:(w00,w01)  [half==1]:(w10,w11)

    const int bbase = blockIdx.x * 16;
    const int tid   = threadIdx.x;

    // ---- Phase 1: per-(p,b) cell index + bilinear weights (one barrier) ----
    for (int q = tid; q < PDIM * 16; q += 256) {
        const int p  = q >> 4;
        const int b  = q & 15;
        const int bb = bbase + b;
        const float x1 = x[(size_t)(2 * p)     * BATCH + bb];
        const float x2 = x[(size_t)(2 * p + 1) * BATCH + bb];

        const float e1 = expf(-fabsf(x1));
        const float c1 = (x1 > 0.f) ? (1.f - 0.5f * e1) : (0.5f * e1);
        int i1 = (int)(c1 * (float)G);
        i1 = min(max(i1, 0), G - 1);
        const float d1 = (x1 - borders[i1]) * invlen[i1];

        const float e2 = expf(-fabsf(x2));
        const float c2 = (x2 > 0.f) ? (1.f - 0.5f * e2) : (0.5f * e2);
        int i2 = (int)(c2 * (float)G);
        i2 = min(max(i2, 0), G - 1);
        const float d2 = (x2 - borders[i2]) * invlen[i2];

        gLDS[p][b]    = i1 * GP1 + i2;
        wLDS[p][0][b] = make_float2((1.f - d1) * (1.f - d2),   // w00 (i1  ,i2  )
                                    (1.f - d1) * d2);          // w01 (i1  ,i2+1)
        wLDS[p][1][b] = make_float2(d1 * (1.f - d2),           // w10 (i1+1,i2  )
                                    d1 * d2);                  // w11 (i1+1,i2+1)
    }
    __syncthreads();

    // ---- Phase 2: barrier-free WMMA accumulation ----
    const int wave  = tid >> 5;
    const int lane  = tid & 31;
    const int half  = lane >> 4;             // 0: rows i1  / 1: rows i1+1
    const int col   = lane & 15;
    const int nbase = wave * 16;

    // per-lane constant part of the gather offset (elements)
    const unsigned laneoff = (unsigned)(nbase + col + half * (GP1 * ODIM));

    v8f acc = {0.f, 0.f, 0.f, 0.f, 0.f, 0.f, 0.f, 0.f};

    for (int p = 0; p < PDIM; ++p) {
        const float* __restrict__ bp = fpT + (size_t)p * ROWS;  // uniform SGPR base

        // Bulk param fetch: 4x ds_load_b128 (cells) + 8x ds_load_b128 (weights)
        int    garr[16];
        float2 warr[16];
        #pragma unroll
        for (int i = 0; i < 4; ++i) {
            int4 t4 = ((const int4*)(&gLDS[p][0]))[i];
            garr[4 * i + 0] = t4.x;  garr[4 * i + 1] = t4.y;
            garr[4 * i + 2] = t4.z;  garr[4 * i + 3] = t4.w;
        }
        #pragma unroll
        for (int i = 0; i < 8; ++i) {
            float4 t4 = ((const float4*)(&wLDS[p][half][0]))[i];
            warr[2 * i + 0] = make_float2(t4.x, t4.y);
            warr[2 * i + 1] = make_float2(t4.z, t4.w);
        }

        #pragma unroll
        for (int b0 = 0; b0 < 16; ++b0) {
            // B gathers: uniform base + 32-bit lane offset, pair via imm offset
            const unsigned off = (unsigned)garr[b0] * ODIM + laneoff;
            const float bv0 = bp[off];           // c00 (h0) / c10 (h1)
            const float bv1 = bp[off + ODIM];    // c01 (h0) / c11 (h1)

            // A: one-hot row m==b0; weights already half-selected, in registers
            const float av0 = (col == b0) ? warr[b0].x : 0.f;
            const float av1 = (col == b0) ? warr[b0].y : 0.f;

            const v2f A  = {av0, av1};
            const v2f Bv = {bv0, bv1};
            acc = __builtin_amdgcn_wmma_f32_16x16x4_f32(
                    false, A, false, Bv, (short)0, acc, false, false);
        }
    }

    // ---- Store: D[m,n] -> out[(nbase+n)*BATCH + bbase + m]
    // lane covers n=col, m = half*8 + j (j=0..7) -> 8 consecutive floats.
    float* op = out + (size_t)(nbase + col) * BATCH + bbase + half * 8;
    float4 lo = make_float4(acc[0], acc[1], acc[2], acc[3]);
    float4 hi = make_float4(acc[4], acc[5], acc[6], acc[7]);
    *(float4*)(op)     = lo;
    *(float4*)(op + 4) = hi;
}

// ---------------------------------------------------------------------------
extern "C" void kernel_launch(void* const* d_in, const int* in_sizes, int n_in,
                              void* d_out, int out_size, void* d_ws, size_t ws_size,
                              hipStream_t stream) {
    const float* x       = (const float*)d_in[0];   // (128, 16384)
    const float* fp      = (const float*)d_in[1];   // (65, 65, 128, 64)
    const float* borders = (const float*)d_in[2];   // (65)
    const float* invlen  = (const float*)d_in[3];   // (64)
    float*       out     = (float*)d_out;           // (128, 16384)
    float*       fpT     = (float*)d_ws;            // 64*4225*128 floats = 138.4 MB

    transpose_fp<<<ROWS / 64, 256, 0, stream>>>(fp, fpT);
    kan2d_wmma<<<BATCH / 16, 256, 0, stream>>>(x, fpT, borders, invlen, out);
}